// Triangles_10754598109362
// MI455X (gfx1250) — compile-verified
//
#include <hip/hip_runtime.h>

typedef __attribute__((ext_vector_type(2))) float v2f;
typedef __attribute__((ext_vector_type(8))) float v8f;

#define P_PTS  131072
#define N_TRI  256

// d_ws / LDS float layout:
//  [0, 10240)      btab : 16 groups x 10 dot-kinds x 32 lanes x float2 (WMMA-B frags)
//  [10240, 12288)  stab : 256 tris x 8 scalars {1/ba2,1/cb2,1/ac2,1/n2, ba2,cb2,ac2, 0}
//  (LDS only) [12288,12672) point staging, [12672,12800) |p|^2 staging
#define BTAB_F 10240
#define STAB_F 2048
#define TBL_F  (BTAB_F + STAB_F)     // 12288 floats = 49152 bytes (12 x 4096)

__device__ __forceinline__ float sgnf(float x) {
    return (x > 0.0f) ? 1.0f : ((x < 0.0f) ? -1.0f : 0.0f);
}

// ---------------------------------------------------------------------------
// Kernel 1: build WMMA-ready tables. For triangle t (group g=t/16, col n0=t%16)
// emit 10 B-matrix columns (vec.xyz in K=0..2, affine offset in K=3):
//  k0..2 : ba|cb|ac       , -dot(e, vertex)        -> D = dot(e, p-V)
//  k3..5 : cross(e,nor)   , -dot(., vertex)        -> sidedness args
//  k6    : nor            , -dot(nor, A)           -> plane distance arg
//  k7..9 : -2A|-2B|-2C    , +|V|^2                 -> |p-V|^2 = pp + D
// B frag layout guess mirrors the documented A 16x4 layout:
//  vgpr0 = {lanes0-15: K=0 , lanes16-31: K=2}, vgpr1 = {K=1 , K=3}.
// ---------------------------------------------------------------------------
__global__ __launch_bounds__(256) void tri_pre_kernel(const float* __restrict__ tri,
                                                      float* __restrict__ tbl)
{
    int t = threadIdx.x;
    if (t >= N_TRI) return;
    const float* v = tri + t * 9;
    float Ax=v[0], Ay=v[1], Az=v[2];
    float Bx=v[3], By=v[4], Bz=v[5];
    float Cx=v[6], Cy=v[7], Cz=v[8];

    float bax=Bx-Ax, bay=By-Ay, baz=Bz-Az;   // ba = B - A
    float cbx=Cx-Bx, cby=Cy-By, cbz=Cz-Bz;   // cb = C - B
    float acx=Ax-Cx, acy=Ay-Cy, acz=Az-Cz;   // ac = A - C

    float nx = bay*acz - baz*acy;            // nor = cross(ba, ac)
    float ny = baz*acx - bax*acz;
    float nz = bax*acy - bay*acx;

    float c1x = bay*nz - baz*ny, c1y = baz*nx - bax*nz, c1z = bax*ny - bay*nx; // cross(ba,nor)
    float c2x = cby*nz - cbz*ny, c2y = cbz*nx - cbx*nz, c2z = cbx*ny - cby*nx; // cross(cb,nor)
    float c3x = acy*nz - acz*ny, c3y = acz*nx - acx*nz, c3z = acx*ny - acy*nx; // cross(ac,nor)

    float ba2 = bax*bax + bay*bay + baz*baz;
    float cb2 = cbx*cbx + cby*cby + cbz*cbz;
    float ac2 = acx*acx + acy*acy + acz*acz;
    float n2  = nx*nx + ny*ny + nz*nz;

    float VX[10] = { bax, cbx, acx, c1x, c2x, c3x, nx, -2.0f*Ax, -2.0f*Bx, -2.0f*Cx };
    float VY[10] = { bay, cby, acy, c1y, c2y, c3y, ny, -2.0f*Ay, -2.0f*By, -2.0f*Cy };
    float VZ[10] = { baz, cbz, acz, c1z, c2z, c3z, nz, -2.0f*Az, -2.0f*Bz, -2.0f*Cz };
    float VW[10] = { -(bax*Ax + bay*Ay + baz*Az),
                     -(cbx*Bx + cby*By + cbz*Bz),
                     -(acx*Cx + acy*Cy + acz*Cz),
                     -(c1x*Ax + c1y*Ay + c1z*Az),
                     -(c2x*Bx + c2y*By + c2z*Bz),
                     -(c3x*Cx + c3y*Cy + c3z*Cz),
                     -(nx*Ax  + ny*Ay  + nz*Az ),
                      (Ax*Ax + Ay*Ay + Az*Az),
                      (Bx*Bx + By*By + Bz*Bz),
                      (Cx*Cx + Cy*Cy + Cz*Cz) };

    int g = t >> 4, n0 = t & 15;
#pragma unroll
    for (int k = 0; k < 10; ++k) {
        float* b = tbl + ((g*10 + k)*32 + n0)*2;
        b[0]  = VX[k];  b[1]  = VY[k];     // lane n0    : K=0, K=1
        b[32] = VZ[k];  b[33] = VW[k];     // lane n0+16 : K=2, K=3
    }
    float* s = tbl + BTAB_F + t*8;
    s[0]=1.0f/ba2; s[1]=1.0f/cb2; s[2]=1.0f/ac2; s[3]=1.0f/n2;
    s[4]=ba2;      s[5]=cb2;      s[6]=ac2;      s[7]=0.0f;
}

// ---------------------------------------------------------------------------
// Kernel 2: 16 points per wave, 128 per block. Tables staged via CDNA5 async
// global->LDS. Per triangle-group of 16: 10x V_WMMA_F32_16X16X4_F32 compute
// all affine dots (D: lane=triangle, vgpr slot=point), then pure elementwise
// math per (point,tri) slot. Butterfly-reduce exp sums across the 16 lanes.
// ---------------------------------------------------------------------------
__global__ __launch_bounds__(256) void tri_sdf_wmma_kernel(const float* __restrict__ p,
                                                           const float* __restrict__ tbl,
                                                           float* __restrict__ out)
{
    __shared__ float smem[TBL_F + 384 + 128];   // 51200 bytes
    const unsigned tid = threadIdx.x;

    // --- async bulk copy of 48KB tables (ASYNCcnt path) -------------------
    {
        unsigned           lds0 = (unsigned)(size_t)(&smem[0]);
        unsigned long long g0   = (unsigned long long)(size_t)tbl;
#pragma unroll
        for (int i = 0; i < (TBL_F * 4) / (256 * 16); ++i) {
            unsigned           lds_addr = lds0 + tid * 16u + (unsigned)i * 4096u;
            unsigned long long gaddr    = g0   + tid * 16u + (unsigned)i * 4096u;
            asm volatile("global_load_async_to_lds_b128 %0, %1, off"
                         :: "v"(lds_addr), "v"(gaddr) : "memory");
        }
    }

    // --- stage this block's 128 points while the async copy is in flight --
    const int blockBase = blockIdx.x * 128;
    float* psm  = smem + TBL_F;          // 384 floats (x,y,z per point)
    float* ppsm = smem + TBL_F + 384;    // 128 floats (|p|^2)
    psm[tid] = p[blockBase * 3 + tid];
    if (tid < 128) psm[256 + tid] = p[blockBase * 3 + 256 + tid];

    asm volatile("s_wait_asynccnt 0x0" ::: "memory");
    __syncthreads();

    if (tid < 128) {
        float x = psm[tid*3], y = psm[tid*3+1], z = psm[tid*3+2];
        ppsm[tid] = x*x + y*y + z*z;
    }
    __syncthreads();

    const int lane = tid & 31, wave = tid >> 5;
    const int m = lane & 15, half = lane >> 4;

    // A fragment: 16x4 = [px py pz 1] rows for this wave's 16 points.
    // vgpr0 = {lanes0-15: K=0(px), lanes16-31: K=2(pz)}, vgpr1 = {K=1(py), K=3(1)}
    const int pl = wave * 16 + m;
    v2f afrag;
    afrag.x = half ? psm[pl*3 + 2] : psm[pl*3 + 0];
    afrag.y = half ? 1.0f          : psm[pl*3 + 1];

    float ppr[8];
#pragma unroll
    for (int r = 0; r < 8; ++r) ppr[r] = ppsm[wave*16 + half*8 + r];

    float acc[8];
#pragma unroll
    for (int r = 0; r < 8; ++r) acc[r] = 0.0f;

    const v2f*   bt = (const v2f*)smem;                 // B frags
    const float4* st = (const float4*)(smem + BTAB_F);  // per-tri scalars
    const v8f cz = {};

    for (int g = 0; g < 16; ++g) {
        const int gb = g * 10 * 32;
        v2f b0 = bt[gb + 0*32 + lane];
        v2f b1 = bt[gb + 1*32 + lane];
        v2f b2 = bt[gb + 2*32 + lane];
        v2f b3 = bt[gb + 3*32 + lane];
        v2f b4 = bt[gb + 4*32 + lane];
        v2f b5 = bt[gb + 5*32 + lane];
        v2f b6 = bt[gb + 6*32 + lane];
        v2f b7 = bt[gb + 7*32 + lane];
        v2f b8 = bt[gb + 8*32 + lane];
        v2f b9 = bt[gb + 9*32 + lane];
        float4 sA = st[(g*16 + m)*2 + 0];   // 1/ba2, 1/cb2, 1/ac2, 1/n2
        float4 sB = st[(g*16 + m)*2 + 1];   // ba2, cb2, ac2, -

        v8f D0 = __builtin_amdgcn_wmma_f32_16x16x4_f32(false, afrag, false, b0, (short)0, cz, false, false);
        v8f D1 = __builtin_amdgcn_wmma_f32_16x16x4_f32(false, afrag, false, b1, (short)0, cz, false, false);
        v8f D2 = __builtin_amdgcn_wmma_f32_16x16x4_f32(false, afrag, false, b2, (short)0, cz, false, false);
        v8f D3 = __builtin_amdgcn_wmma_f32_16x16x4_f32(false, afrag, false, b3, (short)0, cz, false, false);
        v8f D4 = __builtin_amdgcn_wmma_f32_16x16x4_f32(false, afrag, false, b4, (short)0, cz, false, false);
        v8f D5 = __builtin_amdgcn_wmma_f32_16x16x4_f32(false, afrag, false, b5, (short)0, cz, false, false);
        v8f D6 = __builtin_amdgcn_wmma_f32_16x16x4_f32(false, afrag, false, b6, (short)0, cz, false, false);
        v8f D7 = __builtin_amdgcn_wmma_f32_16x16x4_f32(false, afrag, false, b7, (short)0, cz, false, false);
        v8f D8 = __builtin_amdgcn_wmma_f32_16x16x4_f32(false, afrag, false, b8, (short)0, cz, false, false);
        v8f D9 = __builtin_amdgcn_wmma_f32_16x16x4_f32(false, afrag, false, b9, (short)0, cz, false, false);

        // slot r -> point (wave*16 + half*8 + r), triangle (g*16 + m)
#pragma unroll
        for (int r = 0; r < 8; ++r) {
            float u1 = D0[r], u2 = D1[r], u3 = D2[r];
            float sd = sgnf(D3[r]) + sgnf(D4[r]) + sgnf(D5[r]);
            float dn = D6[r];
            float papa = ppr[r] + D7[r];
            float pbpb = ppr[r] + D8[r];
            float pcpc = ppr[r] + D9[r];

            float t1 = fminf(fmaxf(u1 * sA.x, 0.0f), 1.0f);
            float t2 = fminf(fmaxf(u2 * sA.y, 0.0f), 1.0f);
            float t3 = fminf(fmaxf(u3 * sA.z, 0.0f), 1.0f);
            float d1 = fmaf(t1, fmaf(t1, sB.x, -2.0f * u1), papa);
            float d2 = fmaf(t2, fmaf(t2, sB.y, -2.0f * u2), pbpb);
            float d3 = fmaf(t3, fmaf(t3, sB.z, -2.0f * u3), pcpc);

            float dmin = fminf(d1, fminf(d2, d3));
            float dpl  = dn * dn * sA.w;
            float dd   = (sd < 2.0f) ? dmin : dpl;

            float dist = __builtin_amdgcn_sqrtf(fmaxf(dd, 1e-8f)) - 0.04f;
            acc[r] += __builtin_amdgcn_exp2f(dist * -46.166241308446828f);
        }
    }

    // reduce across the 16 triangle-lanes of each half-wave
#pragma unroll
    for (int r = 0; r < 8; ++r) {
        float a = acc[r];
        a += __shfl_xor(a, 1);
        a += __shfl_xor(a, 2);
        a += __shfl_xor(a, 4);
        a += __shfl_xor(a, 8);
        acc[r] = a;
    }

    if (m == 0) {
#pragma unroll
        for (int r = 0; r < 8; ++r) {
            out[blockBase + wave*16 + half*8 + r] =
                -__builtin_amdgcn_logf(fmaxf(acc[r], 1e-6f)) * 0.021660849392498292f;
        }
    }
}

// ---------------------------------------------------------------------------
// Harness entry. d_in[0] = p (131072*3 f32), d_in[1] = points (256*3*3 f32).
// d_ws holds the 48KB WMMA-ready tables.
// ---------------------------------------------------------------------------
extern "C" void kernel_launch(void* const* d_in, const int* in_sizes, int n_in,
                              void* d_out, int out_size, void* d_ws, size_t ws_size,
                              hipStream_t stream)
{
    const float* p    = (const float*)d_in[0];
    const float* tris = (const float*)d_in[1];
    float*       out  = (float*)d_out;
    float*       tbl  = (float*)d_ws;      // TBL_F * 4 = 49152 bytes

    tri_pre_kernel<<<1, 256, 0, stream>>>(tris, tbl);
    tri_sdf_wmma_kernel<<<P_PTS / 128, 256, 0, stream>>>(p, tbl, out);
}